// GCN_90331752169729
// MI455X (gfx1250) — compile-verified
//
#include <hip/hip_runtime.h>

#define K_DIM 500
#define F_DIM 7

typedef __attribute__((ext_vector_type(2))) float v2f;
typedef __attribute__((ext_vector_type(8))) float v8f;

// ---------------------------------------------------------------- utilities
__global__ void gcn_zero_f32(float* __restrict__ p, int n) {
    int i = blockIdx.x * blockDim.x + threadIdx.x;
    if (i < n) p[i] = 0.0f;
}

// WpT[n][k] = (n < 7) ? W[k][n] : 0   -- 16 x 500, zero-padded columns 7..15
__global__ void gcn_build_wpt(const float* __restrict__ W, float* __restrict__ WpT) {
    int tid = blockIdx.x * blockDim.x + threadIdx.x;
    if (tid >= 16 * K_DIM) return;
    int n = tid / K_DIM;
    int k = tid - n * K_DIM;
    WpT[n * K_DIM + k] = (n < F_DIM) ? W[k * F_DIM + n] : 0.0f;
}

// degree counts via float atomics (integer-valued -> exact, order independent)
__global__ void gcn_degrees(const int* __restrict__ src, const int* __restrict__ dst,
                            float* __restrict__ outdeg, float* __restrict__ indeg, int E) {
    int e = blockIdx.x * blockDim.x + threadIdx.x;
    if (e < E) {
        atomicAdd(&outdeg[src[e]], 1.0f);
        atomicAdd(&indeg[dst[e]], 1.0f);
    }
}

__global__ void gcn_norms(const float* __restrict__ outdeg, const float* __restrict__ indeg,
                          float* __restrict__ nout, float* __restrict__ nin, int N) {
    int i = blockIdx.x * blockDim.x + threadIdx.x;
    if (i < N) {
        nout[i] = rsqrtf(fmaxf(outdeg[i], 1.0f));
        nin[i]  = rsqrtf(fmaxf(indeg[i], 1.0f));
    }
}

// feat[N][8] = (h @ W) * norm_out[:,None]  via V_WMMA_F32_16X16X4_F32.
// The per-row norm_out scale commutes with the GEMM, so it is applied once at
// store time instead of on every A operand -> inner loop is pure load + wmma.
// One wave per 16-row tile; 8 waves (256 threads) per block.
__global__ __launch_bounds__(256)
void gcn_gemm_wmma(const float* __restrict__ h, const float* __restrict__ WpT,
                   const float* __restrict__ nout, float* __restrict__ feat, int N) {
    int lane = threadIdx.x & 31;
    int wave = threadIdx.x >> 5;
    int tile = blockIdx.x * 8 + wave;
    int ntiles = (N + 15) >> 4;
    if (tile >= ntiles) return;          // wave-uniform exit: EXEC stays all-ones

    int m0   = tile << 4;
    int half = lane >> 4;                // 0: K-offsets {0,1}; 1: K-offsets {2,3}
    int m    = lane & 15;

    if (m0 + 16 <= N) {
        int row = m0 + m;
        const float* __restrict__ ap = h   + (size_t)row * K_DIM + 2 * half;
        const float* __restrict__ bp = WpT + (size_t)m   * K_DIM + 2 * half;

        v8f acc = {};
        #pragma unroll 5
        for (int k0 = 0; k0 < K_DIM; k0 += 4) {
            v2f a = __builtin_nontemporal_load((const v2f*)(ap + k0)); // h read once
            v2f b = *(const v2f*)(bp + k0);                            // L2-resident
            // 8 args: (neg_a, A, neg_b, B, c_mod, C, reuse_a, reuse_b)
            acc = __builtin_amdgcn_wmma_f32_16x16x4_f32(
                false, a, false, b, (short)0, acc, false, false);
        }
        // D layout: VGPR r -> (M=r, N=lane) lanes 0-15; (M=r+8, N=lane-16) lanes 16-31.
        // Only columns 0..7 are meaningful (col 7 is exactly 0 from the padded B).
        if (m < 8) {
            int mb = m0 + (half ? 8 : 0);
            #pragma unroll
            for (int r = 0; r < 8; ++r)
                feat[(size_t)(mb + r) * 8 + m] = acc[r] * nout[mb + r];
        }
    } else {
        // scalar tail for a partial tile (not hit when N % 16 == 0)
        int row = m0 + lane;
        if (row < N) {
            float s = nout[row];
            const float* hp = h + (size_t)row * K_DIM;
            float accs[F_DIM] = {};
            for (int k = 0; k < K_DIM; ++k) {
                float a = hp[k];
                #pragma unroll
                for (int c = 0; c < F_DIM; ++c)
                    accs[c] += a * WpT[c * K_DIM + k];
            }
            #pragma unroll
            for (int c = 0; c < 8; ++c)
                feat[(size_t)row * 8 + c] = (c < F_DIM) ? accs[c] * s : 0.0f;
        }
    }
}

// agg[dst] += feat[src]; feat (3.2MB) and agg (3.2MB) are L2-resident,
// so the random gathers and float atomics resolve in L2, not HBM.
__global__ void gcn_scatter(const int* __restrict__ src, const int* __restrict__ dst,
                            const float* __restrict__ feat, float* __restrict__ agg, int E) {
    int tid = blockIdx.x * blockDim.x + threadIdx.x;
    int e = tid >> 3;
    int c = tid & 7;
    if (e < E && c < F_DIM) {
        float v = feat[(size_t)src[e] * 8 + c];
        atomicAdd(&agg[(size_t)dst[e] * 8 + c], v);
    }
}

__global__ void gcn_finalize(const float* __restrict__ agg, const float* __restrict__ nin,
                             const float* __restrict__ b, float* __restrict__ out, int N) {
    int tid = blockIdx.x * blockDim.x + threadIdx.x;
    if (tid < N * F_DIM) {
        int n = tid / F_DIM;
        int c = tid - n * F_DIM;
        out[tid] = agg[(size_t)n * 8 + c] * nin[n] + b[c];
    }
}

extern "C" void kernel_launch(void* const* d_in, const int* in_sizes, int n_in,
                              void* d_out, int out_size, void* d_ws, size_t ws_size,
                              hipStream_t stream) {
    const float* h   = (const float*)d_in[0];
    const float* W   = (const float*)d_in[1];
    const float* b   = (const float*)d_in[2];
    const int*   src = (const int*)d_in[3];
    const int*   dst = (const int*)d_in[4];
    float*       out = (float*)d_out;

    const int N = in_sizes[0] / K_DIM;   // 100000
    const int E = in_sizes[3];           // 3300000

    // workspace layout (floats): ~8.03 MB total
    float* ws     = (float*)d_ws;
    float* outdeg = ws;                          // [N]
    float* indeg  = ws + (size_t)N;              // [N]
    float* nout   = ws + 2 * (size_t)N;          // [N]
    float* nin    = ws + 3 * (size_t)N;          // [N]
    float* feat   = ws + 4 * (size_t)N;          // [N][8]
    float* agg    = ws + 12 * (size_t)N;         // [N][8]
    float* WpT    = ws + 20 * (size_t)N;         // [16][500]

    const int T = 256;
    gcn_zero_f32<<<(2 * N + T - 1) / T, T, 0, stream>>>(outdeg, 2 * N);
    gcn_zero_f32<<<(8 * N + T - 1) / T, T, 0, stream>>>(agg, 8 * N);
    gcn_build_wpt<<<(16 * K_DIM + T - 1) / T, T, 0, stream>>>(W, WpT);
    gcn_degrees<<<(E + T - 1) / T, T, 0, stream>>>(src, dst, outdeg, indeg, E);
    gcn_norms<<<(N + T - 1) / T, T, 0, stream>>>(outdeg, indeg, nout, nin, N);

    int ntiles = (N + 15) / 16;
    gcn_gemm_wmma<<<(ntiles + 7) / 8, 256, 0, stream>>>(h, WpT, nout, feat, N);

    long long sthreads = (long long)E * 8;
    gcn_scatter<<<(unsigned)((sthreads + T - 1) / T), T, 0, stream>>>(src, dst, feat, agg, E);

    gcn_finalize<<<(N * F_DIM + T - 1) / T, T, 0, stream>>>(agg, nin, b, out, N);
}